// ARMA_68796786147751
// MI455X (gfx1250) — compile-verified
//
#include <hip/hip_runtime.h>
#include <hip/hip_bf16.h>

typedef __attribute__((ext_vector_type(2))) float v2f;
typedef __attribute__((ext_vector_type(8))) float v8f;

#define NGRAPHS 64
#define HDIM 64

// ---------------- utility ----------------
__global__ void zero_f32(float* __restrict__ p, long n) {
  long i = (long)blockIdx.x * blockDim.x + threadIdx.x;
  if (i < n) p[i] = 0.f;
}

__global__ void deg_kernel(const int* __restrict__ dst, float* __restrict__ deg, int E) {
  int e = blockIdx.x * blockDim.x + threadIdx.x;
  if (e < E) atomicAdd(&deg[dst[e]], 1.0f);
}

__global__ void dinv_kernel(const float* __restrict__ deg, float* __restrict__ dinv, int N) {
  int n = blockIdx.x * blockDim.x + threadIdx.x;
  if (n < N) {
    float d = deg[n];
    dinv[n] = d > 0.f ? rsqrtf(fmaxf(d, 1.f)) : 0.f;
  }
}

__global__ void norm_kernel(const int* __restrict__ src, const int* __restrict__ dst,
                            const float* __restrict__ dinv, float* __restrict__ nrm, int E) {
  int e = blockIdx.x * blockDim.x + threadIdx.x;
  if (e < E) nrm[e] = dinv[src[e]] * dinv[dst[e]];
}

// ---------------- WMMA fp32 GEMM: C[k] = (RELUA? relu(A[k]) : A[k]) @ W[k] (+ bias[k]) --------
// A: [Nrows, Fin] per stack (aStrideK may be 0 to share A), W: [Fin, 64], C: [Nrows, 64]
// block = 128 threads (4 waves), wave w handles cols 16w..16w+15; grid = (Nrows/16, K)
template <int RELUA>
__global__ void gemm_wmma64(const float* __restrict__ A, const float* __restrict__ W,
                            const float* __restrict__ bias, float* __restrict__ C,
                            int Nrows, int Fin,
                            long aStrideK, long wStrideK, long cStrideK, int biasStrideK) {
  int k = blockIdx.y;
  const float* Ak = A + (long)k * aStrideK;
  const float* Wk = W + (long)k * wStrideK;
  float*       Ck = C + (long)k * cStrideK;

  int wave = threadIdx.x >> 5;
  int lane = threadIdx.x & 31;
  int mbase = blockIdx.x << 4;
  int col0 = wave << 4;
  int m = lane & 15;       // row within tile (A) / column (B, C)
  int sel = lane >> 4;     // half-wave select

  int row = mbase + m;
  if (row >= Nrows) row = Nrows - 1;   // clamp: garbage only feeds unstored rows

  const float* arow = Ak + (long)row * Fin + 2 * sel;   // A[m][2*sel], A[m][2*sel+1]

  v8f acc = {};
  for (int k0 = 0; k0 < Fin; k0 += 4) {
    v2f a, b;
    float ax = arow[k0];
    float ay = arow[k0 + 1];
    if (RELUA) { ax = fmaxf(ax, 0.f); ay = fmaxf(ay, 0.f); }
    a.x = ax; a.y = ay;
    int kr = k0 + 2 * sel;
    b.x = Wk[(long)kr * HDIM + col0 + m];          // B[kr][n]
    b.y = Wk[(long)(kr + 1) * HDIM + col0 + m];    // B[kr+1][n]
    acc = __builtin_amdgcn_wmma_f32_16x16x4_f32(false, a, false, b, (short)0, acc,
                                                false, false);
  }

  float bv = 0.f;
  if (bias) bv = bias[(long)k * biasStrideK + col0 + m];

#pragma unroll
  for (int r = 0; r < 8; ++r) {
    int rr = mbase + r + 8 * sel;                  // C row: M = r + 8*(lane>=16)
    if (rr < Nrows)
      Ck[(long)rr * HDIM + col0 + m] = acc[r] + bv;
  }
}

// ---------------- edge scatter: acc[k,dst,:] += norm * out[k,src,:] ----------------
// one thread per (edge, k, 4-float chunk): i in [0, E*32), K=2 stacks, 16 chunks of 4
__global__ void scatter_kernel(const float* __restrict__ out, const int* __restrict__ src,
                               const int* __restrict__ dst, const float* __restrict__ nrm,
                               float* __restrict__ acc, int E, long strideK) {
  long i = (long)blockIdx.x * blockDim.x + threadIdx.x;
  long total = (long)E * 32;
  if (i >= total) return;
  int e = (int)(i >> 5);
  int rem = (int)(i & 31);
  int k = rem >> 4;
  int h0 = (rem & 15) << 2;
  float nm = nrm[e];
  int s = src[e], d = dst[e];
  const float4 v = *(const float4*)(out + (long)k * strideK + (long)s * HDIM + h0);
  float* p = acc + (long)k * strideK + (long)d * HDIM + h0;
  atomicAdd(p + 0, v.x * nm);
  atomicAdd(p + 1, v.y * nm);
  atomicAdd(p + 2, v.z * nm);
  atomicAdd(p + 3, v.w * nm);
}

// ---------------- y = leaky_relu(mean_k relu(acc[k])) ----------------
__global__ void combine_kernel(const float* __restrict__ acc, float* __restrict__ y,
                               long nh4, long strideK) {
  long i = (long)blockIdx.x * blockDim.x + threadIdx.x;
  if (i >= nh4) return;
  const float4 a0 = *(const float4*)(acc + i * 4);
  const float4 a1 = *(const float4*)(acc + strideK + i * 4);
  float4 o;
  o.x = 0.5f * (fmaxf(a0.x, 0.f) + fmaxf(a1.x, 0.f));
  o.y = 0.5f * (fmaxf(a0.y, 0.f) + fmaxf(a1.y, 0.f));
  o.z = 0.5f * (fmaxf(a0.z, 0.f) + fmaxf(a1.z, 0.f));
  o.w = 0.5f * (fmaxf(a0.w, 0.f) + fmaxf(a1.w, 0.f));
  o.x = o.x > 0.f ? o.x : 0.2f * o.x;
  o.y = o.y > 0.f ? o.y : 0.2f * o.y;
  o.z = o.z > 0.f ? o.z : 0.2f * o.z;
  o.w = o.w > 0.f ? o.w : 0.2f * o.w;
  *(float4*)(y + i * 4) = o;
}

// ---------------- mean pool (segment sums via atomics) ----------------
__global__ void pool_kernel(const float* __restrict__ h, const int* __restrict__ batch,
                            float* __restrict__ sums, float* __restrict__ cnt, int N) {
  long i = (long)blockIdx.x * blockDim.x + threadIdx.x;
  if (i >= (long)N * 16) return;
  int n = (int)(i >> 4);
  int c = (int)(i & 15) << 2;
  int g = batch[n];
  const float4 v = *(const float4*)(h + (long)n * HDIM + c);
  float* p = sums + (long)g * HDIM + c;
  atomicAdd(p + 0, v.x);
  atomicAdd(p + 1, v.y);
  atomicAdd(p + 2, v.z);
  atomicAdd(p + 3, v.w);
  if (c == 0) atomicAdd(&cnt[g], 1.f);
}

__global__ void head_kernel(const float* __restrict__ sums, const float* __restrict__ cnt,
                            const float* __restrict__ fcw, const float* __restrict__ fcb,
                            float* __restrict__ out) {
  int g = threadIdx.x;
  if (g < NGRAPHS) {
    float c = fmaxf(cnt[g], 1.f);
    float acc = 0.f;
    for (int hh = 0; hh < HDIM; ++hh) acc += (sums[(long)g * HDIM + hh] / c) * fcw[hh];
    acc += fcb[0];
    out[g] = 1.f / (1.f + __expf(-acc));
  }
}

extern "C" void kernel_launch(void* const* d_in, const int* in_sizes, int n_in,
                              void* d_out, int out_size, void* d_ws, size_t ws_size,
                              hipStream_t stream) {
  const float* x       = (const float*)d_in[0];
  const int*   ei      = (const int*)d_in[1];
  const int*   batch   = (const int*)d_in[2];
  const float* init_w1 = (const float*)d_in[3];
  const float* w1      = (const float*)d_in[4];
  const float* root_w1 = (const float*)d_in[5];
  const float* b1      = (const float*)d_in[6];
  const float* init_w2 = (const float*)d_in[7];
  const float* w2      = (const float*)d_in[8];
  const float* root_w2 = (const float*)d_in[9];
  const float* b2      = (const float*)d_in[10];
  const float* fc_w    = (const float*)d_in[11];
  const float* fc_b    = (const float*)d_in[12];
  float* outp = (float*)d_out;

  const int N = in_sizes[2];
  const int E = in_sizes[1] / 2;
  const int F = in_sizes[0] / N;   // 128
  const int K = 2;                  // stacks (per reference)
  const int* src = ei;
  const int* dst = ei + E;

  // workspace carve-out (floats)
  float* ws   = (float*)d_ws;
  float* deg  = ws;  ws += N;
  float* dinv = ws;  ws += N;
  float* nrm  = ws;  ws += E;
  const long NH = (long)N * HDIM;
  float* bufA = ws;  ws += (long)K * NH;
  float* bufB = ws;  ws += (long)K * NH;
  float* h1   = ws;  ws += NH;
  float* h2   = ws;  ws += NH;
  float* psum = ws;  ws += NGRAPHS * HDIM;
  float* pcnt = ws;  ws += NGRAPHS;

  const int TB = 256;
  const long scatterTotal = (long)E * 32;
  const int scatterBlocks = (int)((scatterTotal + TB - 1) / TB);
  dim3 gGemm((N + 15) / 16, K);

  // ---- GCN norm ----
  zero_f32<<<(N + TB - 1) / TB, TB, 0, stream>>>(deg, N);
  deg_kernel<<<(E + TB - 1) / TB, TB, 0, stream>>>(dst, deg, E);
  dinv_kernel<<<(N + TB - 1) / TB, TB, 0, stream>>>(deg, dinv, N);
  norm_kernel<<<(E + TB - 1) / TB, TB, 0, stream>>>(src, dst, dinv, nrm, E);

  // ---- ARMA layer 1 (Fin = F) ----
  gemm_wmma64<0><<<gGemm, 128, 0, stream>>>(x, init_w1, nullptr, bufA, N, F,
                                            0, (long)F * HDIM, NH, HDIM);
  // t = 0: acc = x@root_w1[0] + b1[0]; scatter adds messages
  gemm_wmma64<0><<<gGemm, 128, 0, stream>>>(x, root_w1, b1, bufB, N, F,
                                            0, (long)F * HDIM, NH, HDIM);
  scatter_kernel<<<scatterBlocks, TB, 0, stream>>>(bufA, src, dst, nrm, bufB, E, NH);
  // t = 1: out = relu(acc)@w1[0]; acc2 = x@root_w1[1] + b1[1]; scatter
  gemm_wmma64<1><<<gGemm, 128, 0, stream>>>(bufB, w1, nullptr, bufA, N, HDIM,
                                            NH, (long)HDIM * HDIM, NH, HDIM);
  gemm_wmma64<0><<<gGemm, 128, 0, stream>>>(x, root_w1 + (long)K * F * HDIM, b1 + K * HDIM,
                                            bufB, N, F, 0, (long)F * HDIM, NH, HDIM);
  scatter_kernel<<<scatterBlocks, TB, 0, stream>>>(bufA, src, dst, nrm, bufB, E, NH);
  combine_kernel<<<(int)((NH / 4 + TB - 1) / TB), TB, 0, stream>>>(bufB, h1, NH / 4, NH);

  // ---- ARMA layer 2 (Fin = HDIM) ----
  gemm_wmma64<0><<<gGemm, 128, 0, stream>>>(h1, init_w2, nullptr, bufA, N, HDIM,
                                            0, (long)HDIM * HDIM, NH, HDIM);
  gemm_wmma64<0><<<gGemm, 128, 0, stream>>>(h1, root_w2, b2, bufB, N, HDIM,
                                            0, (long)HDIM * HDIM, NH, HDIM);
  scatter_kernel<<<scatterBlocks, TB, 0, stream>>>(bufA, src, dst, nrm, bufB, E, NH);
  gemm_wmma64<1><<<gGemm, 128, 0, stream>>>(bufB, w2, nullptr, bufA, N, HDIM,
                                            NH, (long)HDIM * HDIM, NH, HDIM);
  gemm_wmma64<0><<<gGemm, 128, 0, stream>>>(h1, root_w2 + (long)K * HDIM * HDIM, b2 + K * HDIM,
                                            bufB, N, HDIM, 0, (long)HDIM * HDIM, NH, HDIM);
  scatter_kernel<<<scatterBlocks, TB, 0, stream>>>(bufA, src, dst, nrm, bufB, E, NH);
  combine_kernel<<<(int)((NH / 4 + TB - 1) / TB), TB, 0, stream>>>(bufB, h2, NH / 4, NH);

  // ---- mean pool + FC head ----
  zero_f32<<<(NGRAPHS * HDIM + NGRAPHS + TB - 1) / TB, TB, 0, stream>>>(
      psum, NGRAPHS * HDIM + NGRAPHS);  // psum and pcnt are contiguous
  pool_kernel<<<(int)(((long)N * 16 + TB - 1) / TB), TB, 0, stream>>>(h2, batch, psum, pcnt, N);
  head_kernel<<<1, 64, 0, stream>>>(psum, pcnt, fc_w, fc_b, outp);
}